// PolarityGatedAttention_14748917694700
// MI455X (gfx1250) — compile-verified
//
#include <hip/hip_runtime.h>

// PolarityGatedAttention for MI455X (gfx1250), fp32 WMMA path.
// Pipeline: [A] qg/kv GEMMs -> [B] kv-state partials (|k|^p fused) ->
//           [C] q-side attention GEMM (|q|^p fused, S reduced in LDS) ->
//           [D] depthwise 5x5 conv (+= into att) -> [E] gated projection GEMM.

namespace {
constexpr int B_ = 8, N_ = 4096, C_ = 256, H_ = 8, HD = 32;
constexpr int IMG = 64, KW = 5;
constexpr float ALPHA = 4.0f;
constexpr int NCHUNK = 16;
constexpr int CHUNK_N = N_ / NCHUNK; // 256
}

typedef __attribute__((ext_vector_type(2))) float v2f;
typedef __attribute__((ext_vector_type(8))) float v8f;

__device__ __forceinline__ v8f wmma4(v2f a, v2f b, v8f c) {
  // V_WMMA_F32_16X16X4_F32: D = A(16x4) * B(4x16) + C(16x16)
  return __builtin_amdgcn_wmma_f32_16x16x4_f32(
      /*neg_a=*/false, a, /*neg_b=*/false, b,
      /*c_mod=*/(short)0, c, /*reuse_a=*/false, /*reuse_b=*/false);
}

__device__ __forceinline__ float sigmoidf_(float x) {
  return 1.0f / (1.0f + __expf(-x));
}

// relu(x)^p + relu(-x)^p == |x|^p (x==0 -> 0)
__device__ __forceinline__ float spow_(float x, float p) {
  float ax = fabsf(x);
  return ax > 0.0f ? __powf(ax, p) : 0.0f;
}

// ---------------------------------------------------------------------------
// [A] qg & kv GEMMs: x[32768,256] @ {W_qg|W_kv}[256,512] -> q,g,k,v [32768,256]
// Wave = 16 rows x 64 cols (4 accumulators share the A tile).
// grid (256, 16), block 256 (8 waves).
// ---------------------------------------------------------------------------
__global__ void __launch_bounds__(256)
k_gemm_qgkv(const float* __restrict__ x, const float* __restrict__ Wqg,
            const float* __restrict__ Wkv,
            float* __restrict__ qb, float* __restrict__ gb,
            float* __restrict__ kb, float* __restrict__ vb) {
  const int lane = threadIdx.x & 31;
  const int wv   = threadIdx.x >> 5;
  const int hl   = lane >> 4;      // lane half (K phase select)
  const int lm   = lane & 15;
  const int row0 = (blockIdx.x * 8 + wv) * 16;
  const int col0 = blockIdx.y * 64;          // global col in [0,1024)
  const float* W  = (col0 < 512) ? Wqg : Wkv;
  const int    wc = (col0 < 512) ? col0 : (col0 - 512);
  v8f acc0 = {}, acc1 = {}, acc2 = {}, acc3 = {};
  const float* arow = x + (size_t)(row0 + lm) * C_;
#pragma unroll 4
  for (int k0 = 0; k0 < C_; k0 += 4) {
    const int ka = k0 + 2 * hl;              // A: v0<->K{0,2}, v1<->K{1,3}
    v2f a; a.x = arow[ka]; a.y = arow[ka + 1];
    const float* w0 = W + (size_t)ka * (2 * C_) + wc + lm;
    const float* w1 = w0 + 2 * C_;
    v2f b0, b1, b2, b3;
    b0.x = w0[0];  b0.y = w1[0];
    b1.x = w0[16]; b1.y = w1[16];
    b2.x = w0[32]; b2.y = w1[32];
    b3.x = w0[48]; b3.y = w1[48];
    acc0 = wmma4(a, b0, acc0);
    acc1 = wmma4(a, b1, acc1);
    acc2 = wmma4(a, b2, acc2);
    acc3 = wmma4(a, b3, acc3);
  }
  float* dst; int dc;
  if      (col0 < 256) { dst = qb; dc = col0;       }
  else if (col0 < 512) { dst = gb; dc = col0 - 256; }
  else if (col0 < 768) { dst = kb; dc = col0 - 512; }
  else                 { dst = vb; dc = col0 - 768; }
#pragma unroll
  for (int r = 0; r < 8; ++r) {               // C/D: row = r + 8*half, col = lm
    float* o = dst + (size_t)(row0 + r + 8 * hl) * C_ + dc + lm;
    o[0]  = acc0[r];
    o[16] = acc1[r];
    o[32] = acc2[r];
    o[48] = acc3[r];
  }
}

// ---------------------------------------------------------------------------
// [B] kv-state partials: S_chunk[d,e] = sum_{n in chunk} |k|^p[n,d] * v[n,e]
// grid (16 chunks, 64 bh), block 128 (4 waves: 2x2 16x16 tiles of the 32x32 S).
// ---------------------------------------------------------------------------
__global__ void __launch_bounds__(128)
k_kv_state(const float* __restrict__ kb, const float* __restrict__ vb,
           const float* __restrict__ power_p, float* __restrict__ kv_part) {
  const int bh = blockIdx.y, bb = bh >> 3, h = bh & 7;
  const int chunk = blockIdx.x;
  const int lane = threadIdx.x & 31, wv = threadIdx.x >> 5;
  const int hl = lane >> 4, lm = lane & 15;
  const int d0 = (wv >> 1) * 16, e0 = (wv & 1) * 16;
  const float pd = 1.0f + ALPHA * sigmoidf_(power_p[h * HD + d0 + lm]);
  const float* kbase = kb + (size_t)bb * N_ * C_ + h * HD;
  const float* vbase = vb + (size_t)bb * N_ * C_ + h * HD;
  v8f acc = {};
  const int nb = chunk * CHUNK_N;
#pragma unroll 4
  for (int nk = 0; nk < CHUNK_N; nk += 4) {
    const int n = nb + nk + 2 * hl;
    v2f a, b;
    a.x = spow_(kbase[(size_t)n * C_ + d0 + lm], pd);        // A = Kt[d, n]
    a.y = spow_(kbase[(size_t)(n + 1) * C_ + d0 + lm], pd);
    b.x = vbase[(size_t)n * C_ + e0 + lm];                   // B = V[n, e]
    b.y = vbase[(size_t)(n + 1) * C_ + e0 + lm];
    acc = wmma4(a, b, acc);
  }
  float* o = kv_part + ((size_t)chunk * (B_ * H_) + bh) * (HD * HD);
#pragma unroll
  for (int r = 0; r < 8; ++r)
    o[(d0 + r + 8 * hl) * HD + e0 + lm] = acc[r];
}

// ---------------------------------------------------------------------------
// [C] x_att = |q|^p @ S, per (b,h). Reduce the 16 S-partials in LDS first.
// grid (32 n-blocks, 64 bh), block 256 (8 waves x 16 rows; 2 e-tiles each).
// ---------------------------------------------------------------------------
__global__ void __launch_bounds__(256)
k_attn(const float* __restrict__ qb, const float* __restrict__ kv_part,
       const float* __restrict__ power_p, float* __restrict__ att) {
  __shared__ float S[HD * HD];
  __shared__ float pw[HD];
  const int bh = blockIdx.y, bb = bh >> 3, h = bh & 7;
  const int tid = threadIdx.x;
  for (int i = tid; i < HD * HD; i += 256) {
    float s = 0.0f;
#pragma unroll
    for (int c = 0; c < NCHUNK; ++c)
      s += kv_part[((size_t)c * (B_ * H_) + bh) * (HD * HD) + i];
    S[i] = s;
  }
  if (tid < HD) pw[tid] = 1.0f + ALPHA * sigmoidf_(power_p[h * HD + tid]);
  __syncthreads();
  const int lane = tid & 31, wv = tid >> 5;
  const int hl = lane >> 4, lm = lane & 15;
  const int n0 = blockIdx.x * 128 + wv * 16;
  const float* qrow = qb + (size_t)bb * N_ * C_ + (size_t)(n0 + lm) * C_ + h * HD;
  v8f acc0 = {}, acc1 = {};
#pragma unroll
  for (int k0 = 0; k0 < HD; k0 += 4) {
    const int kd = k0 + 2 * hl;
    v2f a, b0, b1;
    a.x = spow_(qrow[kd], pw[kd]);
    a.y = spow_(qrow[kd + 1], pw[kd + 1]);
    b0.x = S[kd * HD + lm];       b0.y = S[(kd + 1) * HD + lm];
    b1.x = S[kd * HD + 16 + lm];  b1.y = S[(kd + 1) * HD + 16 + lm];
    acc0 = wmma4(a, b0, acc0);
    acc1 = wmma4(a, b1, acc1);
  }
  float* ab = att + (size_t)bb * N_ * C_ + h * HD;
#pragma unroll
  for (int r = 0; r < 8; ++r) {
    float* o = ab + (size_t)(n0 + r + 8 * hl) * C_ + lm;
    o[0]  = acc0[r];
    o[16] = acc1[r];
  }
}

// ---------------------------------------------------------------------------
// [D] depthwise 5x5 conv. v_img == flat v buffer viewed as [64, 32, 64, 64].
// Output scatters as att[b, n=y*64+x, cc=ch*8+h2] += conv (after kernel C).
// grid (64 g, 32 ch), block 256; 64x64 plane staged in LDS (16 KB).
// ---------------------------------------------------------------------------
__global__ void __launch_bounds__(256)
k_dwconv(const float* __restrict__ vb, const float* __restrict__ w,
         const float* __restrict__ bias, float* __restrict__ att) {
  __shared__ float plane[IMG * IMG];
  const int g = blockIdx.x, ch = blockIdx.y;
  const float* src = vb + (size_t)g * (HD * IMG * IMG) + (size_t)ch * (IMG * IMG);
  const int tid = threadIdx.x;
  for (int i = tid; i < IMG * IMG; i += 256) plane[i] = src[i];
  __syncthreads();
  float wt[KW * KW];
#pragma unroll
  for (int t = 0; t < KW * KW; ++t) wt[t] = w[ch * KW * KW + t];
  const float bz = bias[ch];
  const int bb = g >> 3, h2 = g & 7, cc = ch * H_ + h2;
  float* ab = att + (size_t)bb * N_ * C_ + cc;
  for (int i = tid; i < IMG * IMG; i += 256) {
    const int y = i >> 6, xx = i & 63;
    float s = bz;
#pragma unroll
    for (int ky = 0; ky < KW; ++ky) {
      const int yy = y + ky - 2;
      if ((unsigned)yy < IMG) {
#pragma unroll
        for (int kx = 0; kx < KW; ++kx) {
          const int xc = xx + kx - 2;
          if ((unsigned)xc < IMG) s += wt[ky * KW + kx] * plane[yy * IMG + xc];
        }
      }
    }
    ab[(size_t)i * C_] += s;
  }
}

// ---------------------------------------------------------------------------
// [E] out = ((att)*g) @ W_proj + b_proj. Gating fused into A-load.
// grid (256, 4), block 256 (8 waves x 16 rows x 64 cols).
// ---------------------------------------------------------------------------
__global__ void __launch_bounds__(256)
k_gemm_proj(const float* __restrict__ att, const float* __restrict__ gb,
            const float* __restrict__ Wp, const float* __restrict__ bp,
            float* __restrict__ out) {
  const int lane = threadIdx.x & 31, wv = threadIdx.x >> 5;
  const int hl = lane >> 4, lm = lane & 15;
  const int row0 = (blockIdx.x * 8 + wv) * 16;
  const int col0 = blockIdx.y * 64;
  v8f acc0 = {}, acc1 = {}, acc2 = {}, acc3 = {};
  const float* arow = att + (size_t)(row0 + lm) * C_;
  const float* grow = gb  + (size_t)(row0 + lm) * C_;
#pragma unroll 4
  for (int k0 = 0; k0 < C_; k0 += 4) {
    const int ka = k0 + 2 * hl;
    v2f a;
    a.x = arow[ka]     * grow[ka];
    a.y = arow[ka + 1] * grow[ka + 1];
    const float* w0 = Wp + (size_t)ka * C_ + col0 + lm;
    const float* w1 = w0 + C_;
    v2f b0, b1, b2, b3;
    b0.x = w0[0];  b0.y = w1[0];
    b1.x = w0[16]; b1.y = w1[16];
    b2.x = w0[32]; b2.y = w1[32];
    b3.x = w0[48]; b3.y = w1[48];
    acc0 = wmma4(a, b0, acc0);
    acc1 = wmma4(a, b1, acc1);
    acc2 = wmma4(a, b2, acc2);
    acc3 = wmma4(a, b3, acc3);
  }
  const float bz0 = bp[col0 + lm],      bz1 = bp[col0 + 16 + lm];
  const float bz2 = bp[col0 + 32 + lm], bz3 = bp[col0 + 48 + lm];
#pragma unroll
  for (int r = 0; r < 8; ++r) {
    float* o = out + (size_t)(row0 + r + 8 * hl) * C_ + col0 + lm;
    o[0]  = acc0[r] + bz0;
    o[16] = acc1[r] + bz1;
    o[32] = acc2[r] + bz2;
    o[48] = acc3[r] + bz3;
  }
}

extern "C" void kernel_launch(void* const* d_in, const int* in_sizes, int n_in,
                              void* d_out, int out_size, void* d_ws, size_t ws_size,
                              hipStream_t stream) {
  const float* x   = (const float*)d_in[0];
  const float* Wqg = (const float*)d_in[1];
  const float* Wkv = (const float*)d_in[2];
  const float* Wp  = (const float*)d_in[3];
  const float* bp  = (const float*)d_in[4];
  const float* dw  = (const float*)d_in[5];
  const float* db  = (const float*)d_in[6];
  const float* pp  = (const float*)d_in[7];
  float* out = (float*)d_out;

  // Workspace layout (floats): q | g | k | v | att | kv_part  (~172 MB)
  const size_t PLANE = (size_t)B_ * N_ * C_; // 8,388,608
  float* ws  = (float*)d_ws;
  float* qb  = ws;
  float* gbuf= qb  + PLANE;
  float* kb  = gbuf+ PLANE;
  float* vb  = kb  + PLANE;
  float* att = vb  + PLANE;
  float* kvp = att + PLANE;                  // NCHUNK*64*1024 floats

  k_gemm_qgkv<<<dim3(256, 16), 256, 0, stream>>>(x, Wqg, Wkv, qb, gbuf, kb, vb);
  k_kv_state <<<dim3(NCHUNK, B_ * H_), 128, 0, stream>>>(kb, vb, pp, kvp);
  k_attn     <<<dim3(N_ / 128, B_ * H_), 256, 0, stream>>>(qb, kvp, pp, att);
  k_dwconv   <<<dim3(B_ * H_, HD), 256, 0, stream>>>(vb, dw, db, att);
  k_gemm_proj<<<dim3(256, C_ / 64), 256, 0, stream>>>(att, gbuf, Wp, bp, out);
}